// Hungarian_48369921687907
// MI455X (gfx1250) — compile-verified
//
#include <hip/hip_runtime.h>
#include <stdint.h>

// ---------------------------------------------------------------------------
// Batched Jonker-Volgenant LSAP on CDNA5 (gfx1250, wave32).
// One wave (32 lanes) per batch sample; 64 columns -> 2 columns per lane.
// Cost tile staged into LDS via Tensor Data Mover; final picked-cost
// reduction via V_WMMA_F32_16X16X4_F32 (B = ones -> row sums).
// ---------------------------------------------------------------------------

typedef float v2f  __attribute__((ext_vector_type(2)));
typedef float v8f  __attribute__((ext_vector_type(8)));
typedef unsigned int u32x4 __attribute__((ext_vector_type(4)));
typedef int   i32x4 __attribute__((ext_vector_type(4)));
typedef int   i32x8 __attribute__((ext_vector_type(8)));

#define NB   64      // N boxes / matrix dim
#define NBAT 256     // batch

__global__ __launch_bounds__(32)
void lsap_kernel(const float* __restrict__ cost,
                 const float* __restrict__ gt,
                 float* __restrict__ out,        // [256*64] ordering (as float)
                 float* __restrict__ ws)         // [256] per-sample picked sums
{
    __shared__ float tile[NB * NB];   // 16 KB cost tile (keep first in LDS)
    __shared__ float u_lds[NB];
    __shared__ float minv_lds[NB];
    __shared__ int   col4row[NB];
    __shared__ int   ord[NB];

    const int b    = blockIdx.x;
    const int lane = threadIdx.x;
    const float INF = __builtin_inff();

    const float* cb = cost + (size_t)b * NB * NB;

    // ---- stage cost[b] (64x64 f32) into LDS --------------------------------
#if __has_builtin(__builtin_amdgcn_tensor_load_to_lds)
    {
        uint64_t ga      = (uint64_t)(uintptr_t)cb;
        uint32_t lds_off = (uint32_t)(uintptr_t)(&tile[0]); // low 32b = LDS offset

        u32x4 g0;
        g0.x = 1u;                                   // count=1, user descriptor
        g0.y = lds_off;                              // lds_addr [63:32]
        g0.z = (uint32_t)(ga & 0xFFFFFFFFu);         // global_addr[31:0]
        g0.w = (uint32_t)((ga >> 32) & 0x01FFFFFFu)  // global_addr[56:32]
               | (2u << 30);                         // type=2 ("image")

        i32x8 g1;
        g1[0] = 0x20000;            // workgroup_mask=0, data_size=2 (4B)
        g1[1] = (int)(64u << 16);   // tensor_dim0[15:0]=64 in bits[63:48]
        g1[2] = (int)(64u << 16);   // dim0 hi=0; tensor_dim1[15:0]=64 in [95:80]
        g1[3] = (int)(64u << 16);   // dim1 hi=0; tile_dim0=64 in [127:112]
        g1[4] = 64;                 // tile_dim1=64, tile_dim2=0
        g1[5] = 64;                 // tensor_dim0_stride lo32 = 64 elements
        g1[6] = 0;                  // stride hi / dim1_stride lo
        g1[7] = 0;

        i32x4 gz = {0, 0, 0, 0};
#if __clang_major__ >= 23
        i32x8 gz8 = {0, 0, 0, 0, 0, 0, 0, 0};
        __builtin_amdgcn_tensor_load_to_lds(g0, g1, gz, gz, gz8, 0);
#else
        __builtin_amdgcn_tensor_load_to_lds(g0, g1, gz, gz, 0);
#endif
        __builtin_amdgcn_s_wait_tensorcnt((short)0);
    }
#else
    for (int k = lane; k < NB * NB / 4; k += 32)
        ((float4*)tile)[k] = ((const float4*)cb)[k];
#endif

    // ---- valid-count: first all-zero [2,3] box -----------------------------
    const float* gb = gt + (size_t)b * NB * 6;
    bool z0 = true, z1 = true;
    #pragma unroll
    for (int t = 0; t < 6; ++t) {
        z0 &= (gb[lane * 6 + t] == 0.0f);
        z1 &= (gb[(lane + 32) * 6 + t] == 0.0f);
    }
    unsigned zm0 = (unsigned)__ballot(z0);
    unsigned zm1 = (unsigned)__ballot(z1);
    int c = zm0 ? __builtin_ctz(zm0) : (zm1 ? 32 + __builtin_ctz(zm1) : NB);

    // ---- init duals / assignment ------------------------------------------
    u_lds[lane] = 0.0f;  u_lds[lane + 32] = 0.0f;
    col4row[lane] = -1;  col4row[lane + 32] = -1;
    float v0 = 0.0f, v1 = 0.0f;     // v[j] for j=lane, j=lane+32
    int   r4c0 = -1, r4c1 = -1;     // row4col[j]
    __syncthreads();

    // ---- JV shortest augmenting path per row ------------------------------
    for (int cur = 0; cur < c; ++cur) {
        float mv0 = INF, mv1 = INF;     // minv[j]
        int   p0 = -1,   p1 = -1;       // path[j]
        bool  rem0 = true, rem1 = true; // remaining[j]
        unsigned long long SR = 0ull;   // scanned rows (uniform)
        int   i = cur;
        float min_val = 0.0f;
        int   sink = -1;

        while (sink < 0) {
            SR |= 1ull << i;
            float ui = u_lds[i];
            const float* rowp = &tile[i * NB];
            float r0 = min_val + rowp[lane]      - ui - v0;
            float r1 = min_val + rowp[lane + 32] - ui - v1;
            if (rem0 && r0 < mv0) { mv0 = r0; p0 = i; }
            if (rem1 && r1 < mv1) { mv1 = r1; p1 = i; }

            // argmin over remaining columns (first-index tie-break like numpy)
            float c0 = rem0 ? mv0 : INF;
            float c1 = rem1 ? mv1 : INF;
            float bv; int bj;
            if (c1 < c0) { bv = c1; bj = lane + 32; }
            else         { bv = c0; bj = lane; }
            #pragma unroll
            for (int off = 16; off; off >>= 1) {
                float ov = __shfl_xor(bv, off);
                int   oj = __shfl_xor(bj, off);
                if (ov < bv || (ov == bv && oj < bj)) { bv = ov; bj = oj; }
            }
            min_val = bv;
            int j = bj;
            int owner = j & 31;
            if (lane == owner) { if (j >= 32) rem1 = false; else rem0 = false; }
            int rj_lo = __shfl(r4c0, owner);
            int rj_hi = __shfl(r4c1, owner);
            int rj = (j >= 32) ? rj_hi : rj_lo;
            if (rj < 0) sink = j; else i = rj;
        }

        // ---- dual updates (before augmentation, as in reference) ----------
        minv_lds[lane] = mv0;  minv_lds[lane + 32] = mv1;
        __syncthreads();
        {
            int r = lane;
            if (((SR >> r) & 1ull) && r != cur)
                u_lds[r] += min_val - minv_lds[col4row[r]];
            r = lane + 32;
            if (((SR >> r) & 1ull) && r != cur)
                u_lds[r] += min_val - minv_lds[col4row[r]];
            if (lane == 0) u_lds[cur] += min_val;
            if (!rem0) v0 -= min_val - mv0;
            if (!rem1) v1 -= min_val - mv1;
        }
        __syncthreads();

        // ---- augment along alternating path -------------------------------
        int j = sink;
        while (true) {
            int own = j & 31;
            int pi_lo = __shfl(p0, own);
            int pi_hi = __shfl(p1, own);
            int i2 = (j >= 32) ? pi_hi : pi_lo;
            if (lane == own) { if (j >= 32) r4c1 = i2; else r4c0 = i2; }
            int nj = col4row[i2];
            __syncthreads();
            if (lane == 0) col4row[i2] = j;
            __syncthreads();
            j = nj;
            if (i2 == cur) break;
        }
    }

    // ---- build ordering: col4row[0:c] ++ sorted unassigned columns --------
    if (lane < c)      ord[lane]      = col4row[lane];
    if (lane + 32 < c) ord[lane + 32] = col4row[lane + 32];
    unsigned um0 = (unsigned)__ballot(r4c0 < 0);
    unsigned um1 = (unsigned)__ballot(r4c1 < 0);
    if (r4c0 < 0) {
        int rank = __builtin_popcount(um0 & ((1u << lane) - 1u));
        ord[c + rank] = lane;
    }
    if (r4c1 < 0) {
        int rank = __builtin_popcount(um0)
                 + __builtin_popcount(um1 & ((1u << lane) - 1u));
        ord[c + rank] = lane + 32;
    }
    __syncthreads();

    // ---- emit ordering (as float, per harness output dtype rule) ----------
    out[(size_t)b * NB + lane]      = (float)ord[lane];
    out[(size_t)b * NB + lane + 32] = (float)ord[lane + 32];

    // ---- picked-cost sum via WMMA: sum_r cost[r][ord[r]] ------------------
    float pa = tile[lane * NB + ord[lane]];
    float pb = tile[(lane + 32) * NB + ord[lane + 32]];
    minv_lds[lane] = pa;  minv_lds[lane + 32] = pb;   // reuse as p[64]
    __syncthreads();

#if __has_builtin(__builtin_amdgcn_wmma_f32_16x16x4_f32)
    {
        // A (16x4 f32): lanes 0-15 hold {K=0,K=1}, lanes 16-31 hold {K=2,K=3}
        int mrow  = lane & 15;
        int kbase = (lane < 16) ? 0 : 2;
        v2f A; A.x = minv_lds[mrow * 4 + kbase];
               A.y = minv_lds[mrow * 4 + kbase + 1];
        v2f Bm; Bm.x = 1.0f; Bm.y = 1.0f;   // all-ones B is layout-invariant
        v8f Cz = {};
        v8f D = __builtin_amdgcn_wmma_f32_16x16x4_f32(
            false, A, false, Bm, (short)0, Cz, false, false);
        // D[m][n] = rowsum(A[m]) for every n; column N=0 lives in lane 0
        // (M=0..7) and lane 16 (M=8..15).
        float s = D[0] + D[1] + D[2] + D[3] + D[4] + D[5] + D[6] + D[7];
        float total = __shfl(s, 0) + __shfl(s, 16);
        if (lane == 0) ws[b] = total;
    }
#else
    {
        float s = pa + pb;
        #pragma unroll
        for (int off = 16; off; off >>= 1) s += __shfl_xor(s, off);
        if (lane == 0) ws[b] = s;
    }
#endif
}

// ---- second kernel: deterministic loss reduction (no atomics) -------------
__global__ __launch_bounds__(256)
void loss_reduce_kernel(const float* __restrict__ ws, float* __restrict__ loss)
{
    __shared__ float sm[256];
    sm[threadIdx.x] = ws[threadIdx.x];
    __syncthreads();
    for (int s = 128; s > 0; s >>= 1) {
        if (threadIdx.x < (unsigned)s) sm[threadIdx.x] += sm[threadIdx.x + s];
        __syncthreads();
    }
    if (threadIdx.x == 0) loss[0] = sm[0] / (float)(NBAT * NB);
}

extern "C" void kernel_launch(void* const* d_in, const int* in_sizes, int n_in,
                              void* d_out, int out_size, void* d_ws, size_t ws_size,
                              hipStream_t stream)
{
    (void)in_sizes; (void)n_in; (void)out_size; (void)ws_size;
    const float* cost = (const float*)d_in[0];
    const float* gt   = (const float*)d_in[1];
    float* out = (float*)d_out;            // [256*64] ordering + [1] loss
    float* ws  = (float*)d_ws;             // [256] per-sample sums

    lsap_kernel<<<NBAT, 32, 0, stream>>>(cost, gt, out, ws);
    loss_reduce_kernel<<<1, 256, 0, stream>>>(ws, out + NBAT * NB);
}